// RSC_52982716563551
// MI455X (gfx1250) — compile-verified
//
#include <hip/hip_runtime.h>

// ---------------------------------------------------------------------------
// Problem constants (from reference):
//   features [B=8, C=1024, T=16, H=14, W=14] fp32
//   W_cls    [C, 400] fp32, labels [B] int, unchanged_mask [B] int(bool)
//   z_grads[b,c,:,:,:] = W_cls[c, labels[b]] / N  (N = 3136) -> constant per (b,c)
//   thr[b] = linear-interp quantile(0.77) over C*N values = order statistic of
//            the 1024 per-channel values (each repeated N times).
//   out[b,c,:] = features[b,c,:] * m[b,c],
//     m = unchanged ? 1 : (w > thr ? 0 : 1/0.77)
// ---------------------------------------------------------------------------

#define B_DIM       8
#define C_DIM       1024
#define NUM_CLASSES 400
#define N_SPATIAL   3136            // 16*14*14
#define N_SPAT4     784             // float4 per channel
#define RETAIN_P    0.77
#define INV_RETAIN  1.2987012987012987f  // 1/0.77

// ---------------------------------------------------------------------------
// Kernel A: per-batch threshold (order-statistic select) + per-(b,c) multiplier.
// One block per batch sample, one thread per channel.
// ---------------------------------------------------------------------------
__global__ __launch_bounds__(C_DIM)
void RSC_thr_mult_kernel(const float* __restrict__ Wcls,
                         const int*  __restrict__ labels,
                         const int*  __restrict__ unchanged,
                         float*      __restrict__ mult) {
    __shared__ float sw[C_DIM];
    __shared__ float s_vlo, s_vhi;

    const int b = blockIdx.x;
    const int c = threadIdx.x;
    const int lab = labels[b];

    // z_grad value for this (b, c): W_cls[c, lab] / N
    const float v = Wcls[c * NUM_CLASSES + lab] * (1.0f / (float)N_SPATIAL);
    sw[c] = v;
    __syncthreads();

    // Stable rank: unique in [0, C) even with ties.
    int r = 0;
    #pragma unroll 4
    for (int j = 0; j < C_DIM; ++j) {
        const float w = sw[j];
        r += (int)((w < v) || (w == v && j < c));
    }

    // numpy linear-interpolation quantile over the *expanded* C*N multiset.
    const long long M    = (long long)C_DIM * N_SPATIAL;
    const double    pos  = RETAIN_P * (double)(M - 1);
    const long long i_lo = (long long)pos;               // floor (pos > 0)
    const double    frac = pos - (double)i_lo;
    const long long i_hi = (i_lo + 1 < M) ? (i_lo + 1) : (M - 1);
    const int b_lo = (int)(i_lo / N_SPATIAL);            // distinct-value block
    const int b_hi = (int)(i_hi / N_SPATIAL);

    if (r == b_lo) s_vlo = v;
    if (r == b_hi) s_vhi = v;
    __syncthreads();

    const float thr = (float)((double)s_vlo + frac * ((double)s_vhi - (double)s_vlo));

    float m;
    if (unchanged[b] != 0)  m = 1.0f;                    // keep features verbatim
    else if (v > thr)       m = 0.0f;                    // zero_mask channel
    else                    m = INV_RETAIN;              // retained, rescaled
    mult[b * C_DIM + c] = m;
}

// ---------------------------------------------------------------------------
// Kernel B: streaming scale. 3136 blocks x 256 threads x 8 float4 each
// == 6,422,528 float4 == 25,690,112 floats (exact tile, no bounds checks,
// EXEC all-ones for every async op).
// CDNA5 path: global_load_async_to_lds_b128 (ASYNCcnt) -> s_wait_asynccnt 0
// -> ds_load_b128 -> scale -> global_store_b128.
// ---------------------------------------------------------------------------
__global__ __launch_bounds__(256)
void RSC_scale_kernel(const float* __restrict__ feat,
                      const float* __restrict__ mult,
                      float*       __restrict__ out) {
    __shared__ float4 tile[2048];                        // 32 KB staging
    const unsigned t    = threadIdx.x;
    const unsigned base = blockIdx.x * 2048u;            // float4 index

    const float4* __restrict__ f4 = (const float4*)feat;
    float4*       __restrict__ o4 = (float4*)out;

    const unsigned lds_base = (unsigned)(size_t)(&tile[0]);

    // Issue 8 back-to-back async 128b global->LDS copies (deep MLP, no VGPR
    // landing registers; tracked on ASYNCcnt).
    #pragma unroll
    for (int j = 0; j < 8; ++j) {
        const unsigned idx4 = base + (unsigned)j * 256u + t;
        const float4*  gp   = f4 + idx4;
        const unsigned lds  = lds_base + ((unsigned)j * 256u + t) * 16u;
        asm volatile("global_load_async_to_lds_b128 %0, %1, off"
                     :
                     : "v"(lds), "v"(gp)
                     : "memory");
    }
    // Wait for all async copies from this wave to land in LDS.
    asm volatile("s_wait_asynccnt 0" ::: "memory");

    // Read back own slots, apply per-channel multiplier, store.
    #pragma unroll
    for (int j = 0; j < 8; ++j) {
        const unsigned idx4 = base + (unsigned)j * 256u + t;
        const float    m    = mult[idx4 / (unsigned)N_SPAT4];  // flat (b*C+c)
        float4 v = tile[(unsigned)j * 256u + t];
        v.x *= m; v.y *= m; v.z *= m; v.w *= m;
        o4[idx4] = v;
    }
}

// ---------------------------------------------------------------------------
// Host-side launcher.
// d_in[0]=features(f32), d_in[1]=W_cls(f32), d_in[2]=labels(i32),
// d_in[3]=unchanged_mask(i32). d_out = f32 [8,1024,16,14,14].
// d_ws: first 8192 floats hold per-(b,c) multipliers.
// ---------------------------------------------------------------------------
extern "C" void kernel_launch(void* const* d_in, const int* in_sizes, int n_in,
                              void* d_out, int out_size, void* d_ws, size_t ws_size,
                              hipStream_t stream) {
    (void)in_sizes; (void)n_in; (void)out_size; (void)ws_size;

    const float* feat      = (const float*)d_in[0];
    const float* Wcls      = (const float*)d_in[1];
    const int*   labels    = (const int*)d_in[2];
    const int*   unchanged = (const int*)d_in[3];
    float*       out       = (float*)d_out;
    float*       mult      = (float*)d_ws;               // 8192 floats

    RSC_thr_mult_kernel<<<B_DIM, C_DIM, 0, stream>>>(Wcls, labels, unchanged, mult);

    // total float4 = 8*1024*3136/4 = 6,422,528 = 3136 blocks * 256 thr * 8
    RSC_scale_kernel<<<3136, 256, 0, stream>>>(feat, mult, out);
}